// DFPSSampler_23845658427862
// MI455X (gfx1250) — compile-verified
//
#include <hip/hip_runtime.h>
#include <stdint.h>

// Problem shape (from setup_inputs)
#define BATCH   8
#define NPTS    131072
#define NBLK    16                  // workgroups per batch
#define PTS     (NPTS / NBLK)       // 8192 points per workgroup
#define THREADS 1024
#define PPT     (PTS / THREADS)     // 8 points per thread (xyz + min-dist in VGPRs)
#define INIT_DIST 1e10f

typedef unsigned int u32;
typedef unsigned int u32x4 __attribute__((ext_vector_type(4)));
typedef int          i32x8 __attribute__((ext_vector_type(8)));
typedef int          i32x4 __attribute__((ext_vector_type(4)));

// Workspace: per-batch arrival counter + double-buffered per-block candidates.
struct WS {
    unsigned int arrive[BATCH];
    float candV[2][BATCH][NBLK];
    int   candI[2][BATCH][NBLK];
    float candX[2][BATCH][NBLK];
    float candY[2][BATCH][NBLK];
    float candZ[2][BATCH][NBLK];
};

__global__ void fps_init_kernel(WS* ws) {
    int t = threadIdx.x;
    if (t < BATCH) ws->arrive[t] = 0u;
}

__global__ __launch_bounds__(THREADS)
void fps_kernel(const float* __restrict__ points, int* __restrict__ out,
                WS* __restrict__ ws, int npoint)
{
    __shared__ float spts[PTS * 3];          // 96 KB xyz shard (AoS), for winner lookup
    __shared__ float redV[THREADS / 32];
    __shared__ int   redI[THREADS / 32];
    __shared__ float bq[3];

    const int b    = blockIdx.x / NBLK;
    const int blk  = blockIdx.x % NBLK;
    const int tid  = threadIdx.x;
    const int lane = tid & 31;
    const int wave = tid >> 5;
    const int baseIdx = blk * PTS;
    const float* src = points + ((size_t)b * NPTS + baseIdx) * 3;

    // ---- One-time stage: Tensor Data Mover bulk DMA global -> LDS (wave 0 only;
    //      TDM ignores EXEC, so exactly one wave issues the descriptor). ----
    if (wave == 0) {
        const u32 elems = (u32)(PTS * 3);                 // 24576 dwords, contiguous 1-D tile
        u32 lds_off = (u32)(uintptr_t)(&spts[0]);         // low 32 bits of LDS aperture addr = offset
        uint64_t ga = (uint64_t)(uintptr_t)src;
        u32x4 g0;
        g0[0] = 1u;                                       // count=1 (valid user descriptor)
        g0[1] = lds_off;                                  // lds_addr
        g0[2] = (u32)(ga & 0xFFFFFFFFu);                  // global_addr[31:0]
        g0[3] = (u32)((ga >> 32) & 0x1FFFFFFu) | (2u << 30); // global_addr[56:32] | type=2
        i32x8 g1;
        g1[0] = (int)(2u << 16);                          // data_size = 4 bytes
        g1[1] = (int)((elems & 0xFFFFu) << 16);           // tensor_dim0[15:0]
        g1[2] = (int)(((elems >> 16) & 0xFFFFu) | (1u << 16)); // tensor_dim0[31:16], tensor_dim1=1
        g1[3] = (int)((elems & 0xFFFFu) << 16);           // tile_dim0 = 24576
        g1[4] = 1;                                        // tile_dim1 = 1
        g1[5] = (int)elems;                               // tensor_dim0_stride
        g1[6] = 0;
        g1[7] = 0;
        i32x4 gz4 = (i32x4){0, 0, 0, 0};
        i32x8 gz8;
        gz8[0] = 0; gz8[1] = 0; gz8[2] = 0; gz8[3] = 0;
        gz8[4] = 0; gz8[5] = 0; gz8[6] = 0; gz8[7] = 0;
        __builtin_amdgcn_tensor_load_to_lds(g0, g1, gz4, gz4, gz8, 0);
        __builtin_amdgcn_s_wait_tensorcnt(0);
    }
    __syncthreads();                                      // all waves see DMA'd LDS

    // ---- One-time hoist: this thread's 8 consecutive points -> VGPRs via
    //      six aligned ds_load_b128 (96 B/thread). Coordinates never change,
    //      so the per-iteration body touches no memory for points at all. ----
    float px[PPT], py[PPT], pz[PPT];
    {
        const float4* b4 = (const float4*)&spts[tid * (PPT * 3)];
        float t[PPT * 3];
#pragma unroll
        for (int j = 0; j < (PPT * 3) / 4; ++j) {
            float4 v = b4[j];
            t[4 * j + 0] = v.x; t[4 * j + 1] = v.y;
            t[4 * j + 2] = v.z; t[4 * j + 3] = v.w;
        }
#pragma unroll
        for (int p = 0; p < PPT; ++p) {
            px[p] = t[3 * p + 0];
            py[p] = t[3 * p + 1];
            pz[p] = t[3 * p + 2];
        }
    }

    // Per-thread min distances held entirely in VGPRs.
    float md[PPT];
#pragma unroll
    for (int p = 0; p < PPT; ++p) md[p] = INIT_DIST;

    // First query = point 0 of the batch (first selected index is 0).
    float qx = points[(size_t)b * NPTS * 3 + 0];
    float qy = points[(size_t)b * NPTS * 3 + 1];
    float qz = points[(size_t)b * NPTS * 3 + 2];
    if (blk == 0 && tid == 0) out[b * npoint + 0] = 0;

    const int liBase = tid * PPT;                         // consecutive ownership

    for (int k = 0; k < npoint - 1; ++k) {
        // 1) Pure-VALU min-dist update + per-thread argmax (ascending index
        //    keeps earliest on ties, matching jnp.argmax first-occurrence).
        float bestV = -1.0f;
        int   bestI = 0x7FFFFFFF;
#pragma unroll
        for (int p = 0; p < PPT; ++p) {
            float dx = px[p] - qx, dy = py[p] - qy, dz = pz[p] - qz;
            float d = dx * dx + dy * dy + dz * dz;
            float m = md[p] < d ? md[p] : d;
            md[p] = m;
            if (m > bestV) { bestV = m; bestI = liBase + p; }
        }
        // 2) wave32 shuffle reduction
        for (int off = 16; off > 0; off >>= 1) {
            float ov = __shfl_down(bestV, off, 32);
            int   oi = __shfl_down(bestI, off, 32);
            if (ov > bestV || (ov == bestV && oi < bestI)) { bestV = ov; bestI = oi; }
        }
        if (lane == 0) { redV[wave] = bestV; redI[wave] = bestI; }
        __syncthreads();

        const int par = k & 1;
        if (wave == 0) {
            // 3) Block-level reduce (32 wave leaders == one wave32).
            float v = redV[lane];
            int   i = redI[lane];
            for (int off = 16; off > 0; off >>= 1) {
                float ov = __shfl_down(v, off, 32);
                int   oi = __shfl_down(i, off, 32);
                if (ov > v || (ov == v && oi < i)) { v = ov; i = oi; }
            }
            // 4) Publish candidate (value, global idx, xyz) and arrive.
            if (lane == 0) {
                ws->candV[par][b][blk] = v;
                ws->candI[par][b][blk] = baseIdx + i;
                ws->candX[par][b][blk] = spts[i * 3 + 0];
                ws->candY[par][b][blk] = spts[i * 3 + 1];
                ws->candZ[par][b][blk] = spts[i * 3 + 2];
                __threadfence();
                __hip_atomic_fetch_add(&ws->arrive[b], 1u,
                                       __ATOMIC_RELEASE, __HIP_MEMORY_SCOPE_AGENT);
                const unsigned target = (unsigned)(k + 1) * NBLK;
                while (__hip_atomic_load(&ws->arrive[b],
                                         __ATOMIC_ACQUIRE, __HIP_MEMORY_SCOPE_AGENT) < target)
                    __builtin_amdgcn_s_sleep(1);
            }
            // 5) Cross-block argmax: 16 candidates (xyz rides along -> no 2nd sync).
            float cv = -1.0f; int ci = 0x7FFFFFFF;
            float cx = 0.f, cy = 0.f, cz = 0.f;
            if (lane < NBLK) {
                cv = ws->candV[par][b][lane];
                ci = ws->candI[par][b][lane];
                cx = ws->candX[par][b][lane];
                cy = ws->candY[par][b][lane];
                cz = ws->candZ[par][b][lane];
            }
            for (int off = 8; off > 0; off >>= 1) {
                float ov = __shfl_down(cv, off, 32);
                int   oi = __shfl_down(ci, off, 32);
                float ox = __shfl_down(cx, off, 32);
                float oy = __shfl_down(cy, off, 32);
                float oz = __shfl_down(cz, off, 32);
                if (ov > cv || (ov == cv && oi < ci)) { cv = ov; ci = oi; cx = ox; cy = oy; cz = oz; }
            }
            if (lane == 0) {
                bq[0] = cx; bq[1] = cy; bq[2] = cz;
                if (blk == 0) out[b * npoint + (k + 1)] = ci;
            }
        }
        __syncthreads();
        qx = bq[0]; qy = bq[1]; qz = bq[2];
    }
}

extern "C" void kernel_launch(void* const* d_in, const int* in_sizes, int n_in,
                              void* d_out, int out_size, void* d_ws, size_t ws_size,
                              hipStream_t stream) {
    const float* points = (const float*)d_in[0];   // [B, N, 3] float32
    // d_in[1] = features (unused by D-FPS), d_in[2] = npoint scalar
    int* out = (int*)d_out;                        // [B, npoint] int32 indices
    WS* ws = (WS*)d_ws;
    const int npoint = out_size / BATCH;           // 1024

    fps_init_kernel<<<1, 32, 0, stream>>>(ws);     // deterministic counter reset each call
    fps_kernel<<<dim3(BATCH * NBLK), dim3(THREADS), 0, stream>>>(points, out, ws, npoint);
}